// WirelessSplitNet_78512002171344
// MI455X (gfx1250) — compile-verified
//
#include <hip/hip_runtime.h>
#include <hip/hip_bf16.h>

typedef float v2f __attribute__((ext_vector_type(2)));
typedef float v8f __attribute__((ext_vector_type(8)));

constexpr int kB    = 1000;
constexpr int kD    = 8;
constexpr int kPRE  = 98;
constexpr int kNEXT = 512;
constexpr int kSC   = 32;
constexpr int kANT  = 5;

constexpr int AS_STRIDE = 104;   // 98 padded to 100 used, stride 104
constexpr int HS_STRIDE = 520;   // 512 + pad

// ---------------------------------------------------------------------------
// Kernel 0: zero `received` accumulator, compute coef[d,c] = b_mat * h_vec
// ---------------------------------------------------------------------------
__global__ __launch_bounds__(256) void k_init(
    const float* __restrict__ b_mat, const float* __restrict__ h_mat,
    const float* __restrict__ a_mat, const int* __restrict__ k_idx,
    float* __restrict__ received, float* __restrict__ coef) {
  int i = blockIdx.x * 256 + threadIdx.x;
  if (i < kB * kSC) received[i] = 0.0f;
  if (i < kD * kSC) {
    int d = i >> 5, c = i & 31;
    int ksel = k_idx[c];
    float s = 0.0f;
    for (int a = 0; a < kANT; ++a)
      s += h_mat[(d * kSC + ksel) * kANT + a] * a_mat[c * kANT + a];
    coef[i] = b_mat[i] * s;
  }
}

// ---------------------------------------------------------------------------
// Kernel 1: per (d, 16-row tile of B):
//   H = relu(X @ sub_w + sub_b) ; LN ; Cut = relu(H @ cut_w + cut_b)
//   received[b,c] += Cut[b,c] * coef[d,c]     (float atomics over d)
// All GEMMs via V_WMMA_F32_16X16X4_F32 (full fp32 precision).
// ---------------------------------------------------------------------------
__global__ __launch_bounds__(128) void k_main(
    const float* __restrict__ x,     const float* __restrict__ sub_w,
    const float* __restrict__ sub_b, const float* __restrict__ ln_g,
    const float* __restrict__ ln_b,  const float* __restrict__ cut_w,
    const float* __restrict__ cut_b, const float* __restrict__ coef,
    float* __restrict__ received) {
  __shared__ float As[16 * AS_STRIDE];
  __shared__ float Hs[16 * HS_STRIDE];
  __shared__ float red1[16][8];
  __shared__ float red2[16][8];
  __shared__ float mu_s[16];
  __shared__ float inv_s[16];

  const int tid  = threadIdx.x;
  const int w    = tid >> 5;        // wave id 0..3
  const int lane = tid & 31;
  const int half = lane >> 4;       // 0: lanes 0-15, 1: lanes 16-31
  const int lr   = lane & 15;
  const int d    = blockIdx.y;
  const int b0   = blockIdx.x * 16;

  // ---- load 16x98 A tile (zero pad K to 100+, pad short rows) ----
  for (int idx = tid; idx < 16 * AS_STRIDE; idx += 128) {
    int r = idx / AS_STRIDE, c = idx - r * AS_STRIDE;
    int b = b0 + r;
    float v = 0.0f;
    if (c < kPRE && b < kB) v = x[b * (kD * kPRE) + d * kPRE + c];
    As[idx] = v;
  }
  __syncthreads();

  // ---- GEMM1: (16 x 98) @ (98 x 512), wave w owns cols [w*128, w*128+128) ----
  v8f acc[8];
#pragma unroll
  for (int t = 0; t < 8; ++t)
#pragma unroll
    for (int r = 0; r < 8; ++r) acc[t][r] = 0.0f;

  const float* W = sub_w + d * kPRE * kNEXT;
  for (int kk = 0; kk < 100; kk += 4) {
    int k0 = kk + 2 * half;                 // this half-wave's K pair
    v2f a;
    a.x = As[lr * AS_STRIDE + k0];          // zero beyond K=97
    a.y = As[lr * AS_STRIDE + k0 + 1];
    int kc = (k0 <= 96) ? k0 : 96;          // clamp B row (A is 0 there)
#pragma unroll
    for (int t = 0; t < 8; ++t) {
      int n = w * 128 + t * 16 + lr;
      v2f bf;
      bf.x = W[kc * kNEXT + n];
      bf.y = W[(kc + 1) * kNEXT + n];
      acc[t] = __builtin_amdgcn_wmma_f32_16x16x4_f32(
          false, a, false, bf, (short)0, acc[t], false, false);
    }
  }

  // ---- epilogue 1: + sub_b, relu, to LDS ----
  const float* Bv = sub_b + d * kNEXT;
#pragma unroll
  for (int t = 0; t < 8; ++t) {
    int n = w * 128 + t * 16 + lr;
    float bias = Bv[n];
#pragma unroll
    for (int r = 0; r < 8; ++r) {
      int m = r + 8 * half;
      Hs[m * HS_STRIDE + n] = fmaxf(acc[t][r] + bias, 0.0f);
    }
  }
  __syncthreads();

  // ---- LayerNorm over 512 per row ----
  {
    int row = tid >> 3, sub = tid & 7;
    float s = 0.0f, s2 = 0.0f;
    for (int c = sub * 64; c < sub * 64 + 64; ++c) {
      float v = Hs[row * HS_STRIDE + c];
      s += v;
      s2 += v * v;
    }
    red1[row][sub] = s;
    red2[row][sub] = s2;
  }
  __syncthreads();
  if (tid < 16) {
    float s = 0.0f, s2 = 0.0f;
    for (int j = 0; j < 8; ++j) { s += red1[tid][j]; s2 += red2[tid][j]; }
    float mu  = s * (1.0f / 512.0f);
    float var = s2 * (1.0f / 512.0f) - mu * mu;
    mu_s[tid]  = mu;
    inv_s[tid] = rsqrtf(var + 1e-5f);
  }
  __syncthreads();
  {
    const float* G  = ln_g + d * kNEXT;
    const float* Be = ln_b + d * kNEXT;
    for (int idx = tid; idx < 16 * 512; idx += 128) {
      int row = idx >> 9, col = idx & 511;
      float v = Hs[row * HS_STRIDE + col];
      Hs[row * HS_STRIDE + col] =
          (v - mu_s[row]) * inv_s[row] * G[col] + Be[col];
    }
  }
  __syncthreads();

  // ---- GEMM2: (16 x 512) @ (512 x 32), waves 0,1 own 16 cols each ----
  if (w < 2) {
    const float* Cw = cut_w + d * kNEXT * kSC;
    int n0 = w * 16;
    v8f acc2;
#pragma unroll
    for (int r = 0; r < 8; ++r) acc2[r] = 0.0f;
    for (int kk = 0; kk < 512; kk += 4) {
      int k0 = kk + 2 * half;
      v2f a;
      a.x = Hs[lr * HS_STRIDE + k0];
      a.y = Hs[lr * HS_STRIDE + k0 + 1];
      v2f bf;
      bf.x = Cw[k0 * kSC + n0 + lr];
      bf.y = Cw[(k0 + 1) * kSC + n0 + lr];
      acc2 = __builtin_amdgcn_wmma_f32_16x16x4_f32(
          false, a, false, bf, (short)0, acc2, false, false);
    }
    int c = n0 + lr;
    float cb = cut_b[d * kSC + c];
    float cf = coef[d * kSC + c];
#pragma unroll
    for (int r = 0; r < 8; ++r) {
      int b = b0 + r + 8 * half;
      if (b < kB) {
        float v = fmaxf(acc2[r] + cb, 0.0f);
        atomicAdd(&received[b * kSC + c], v * cf);
      }
    }
  }
}

// ---------------------------------------------------------------------------
// Kernel 2: tail MLP per row (one wave32 per row, 4 rows per block)
// ---------------------------------------------------------------------------
__global__ __launch_bounds__(128) void k_tail(
    const float* __restrict__ received, const float* __restrict__ noise,
    const float* __restrict__ a_mat,    const float* __restrict__ clb,
    const float* __restrict__ fc3_w,    const float* __restrict__ fc3_b,
    const float* __restrict__ fc4_w,    const float* __restrict__ fc4_b,
    const float* __restrict__ out_w,    const float* __restrict__ out_b,
    float* __restrict__ out) {
  __shared__ float z[4][32];
  __shared__ float z3[4][98];
  __shared__ float z4[4][49];
  __shared__ float lg[4][10];
  __shared__ float lse[4];

  const int tid = threadIdx.x;
  const int g = tid >> 5, lane = tid & 31;
  const int b = blockIdx.x * 4 + g;   // 250 * 4 == 1000 exactly

  // z = relu(received + scaled_noise + cut_layer_bias)
  {
    float sn = 0.0f;
    for (int a = 0; a < kANT; ++a)
      sn += noise[(b * kSC + lane) * kANT + a] * a_mat[lane * kANT + a];
    z[g][lane] = fmaxf(received[b * kSC + lane] + sn + clb[lane], 0.0f);
  }
  __syncthreads();

  for (int k = lane; k < 98; k += 32) {
    float s = fc3_b[k];
    for (int j = 0; j < 32; ++j) s += z[g][j] * fc3_w[j * 98 + k];
    z3[g][k] = fmaxf(s, 0.0f);
  }
  __syncthreads();

  for (int k = lane; k < 49; k += 32) {
    float s = fc4_b[k];
    for (int j = 0; j < 98; ++j) s += z3[g][j] * fc4_w[j * 49 + k];
    z4[g][k] = fmaxf(s, 0.0f);
  }
  __syncthreads();

  for (int k = lane; k < 10; k += 32) {
    float s = out_b[k];
    for (int j = 0; j < 49; ++j) s += z4[g][j] * out_w[j * 10 + k];
    lg[g][k] = s;
  }
  __syncthreads();

  if (lane == 0) {
    float m = lg[g][0];
    for (int k = 1; k < 10; ++k) m = fmaxf(m, lg[g][k]);
    float s = 0.0f;
    for (int k = 0; k < 10; ++k) s += __expf(lg[g][k] - m);
    lse[g] = m + __logf(s);
  }
  __syncthreads();

  if (lane < 10) out[b * 10 + lane] = lg[g][lane] - lse[g];
}

// ---------------------------------------------------------------------------
extern "C" void kernel_launch(void* const* d_in, const int* in_sizes, int n_in,
                              void* d_out, int out_size, void* d_ws,
                              size_t ws_size, hipStream_t stream) {
  const float* x      = (const float*)d_in[0];
  const float* noise  = (const float*)d_in[1];
  const float* sub_w  = (const float*)d_in[2];
  const float* sub_b  = (const float*)d_in[3];
  const float* ln_g   = (const float*)d_in[4];
  const float* ln_b   = (const float*)d_in[5];
  const float* cut_w  = (const float*)d_in[6];
  const float* cut_b  = (const float*)d_in[7];
  const float* b_mat  = (const float*)d_in[8];
  const float* h_mat  = (const float*)d_in[9];
  const float* a_mat  = (const float*)d_in[10];
  const float* clb    = (const float*)d_in[11];
  const float* fc3_w  = (const float*)d_in[12];
  const float* fc3_b  = (const float*)d_in[13];
  const float* fc4_w  = (const float*)d_in[14];
  const float* fc4_b  = (const float*)d_in[15];
  const float* out_w  = (const float*)d_in[16];
  const float* out_b  = (const float*)d_in[17];
  const int*   k_idx  = (const int*)d_in[18];

  float* received = (float*)d_ws;                  // kB*kSC floats
  float* coef     = received + kB * kSC;           // kD*kSC floats

  k_init<<<(kB * kSC + 255) / 256, 256, 0, stream>>>(b_mat, h_mat, a_mat,
                                                     k_idx, received, coef);

  dim3 grid1((kB + 15) / 16, kD);
  k_main<<<grid1, 128, 0, stream>>>(x, sub_w, sub_b, ln_g, ln_b, cut_w, cut_b,
                                    coef, received);

  k_tail<<<kB / 4, 128, 0, stream>>>(received, noise, a_mat, clb, fc3_w, fc3_b,
                                     fc4_w, fc4_b, out_w, out_b, (float*)d_out);
}